// WindowCrossAttention_82429012345311
// MI455X (gfx1250) — compile-verified
//
#include <hip/hip_runtime.h>
#include <hip/hip_bf16.h>
#include <cmath>

typedef float v2f __attribute__((ext_vector_type(2)));
typedef float v8f __attribute__((ext_vector_type(8)));

// Problem constants (fixed by reference setup_inputs)
#define BATCH   2
#define NQ      512
#define DIM     512
#define NHEADS  8
#define HD      64
#define NTOK    87040
#define NKEYS   512           // 4 levels * 128 window positions
#define SSTRIDE 520           // LDS score row stride (floats), even -> float2 aligned

// ---------------------------------------------------------------------------
// Zero fill
// ---------------------------------------------------------------------------
__global__ void fill_zero_kernel(float* __restrict__ p, int n) {
    int i = blockIdx.x * blockDim.x + threadIdx.x;
    if (i < n) p[i] = 0.0f;
}

// ---------------------------------------------------------------------------
// Generic fp32 WMMA GEMM:  C[M,N] = A[M,K] @ (TRANSB ? B^T : B) + bias
// Tile: workgroup = 8 waves, 128 (M) x 64 (N); wave owns 16 rows x 64 cols.
// Requires M % 128 == 0, N % 64 == 0, K % 8 == 0 (true for all uses here).
// TRANSB is a compile-time template arg -> straight-line inner loop.
// ---------------------------------------------------------------------------
template <int TRANSB>
__global__ void gemm_wmma_f32(const float* __restrict__ A, int lda,
                              const float* __restrict__ Bm, int ldb,
                              const float* __restrict__ bias,
                              float* __restrict__ C, int ldc, int K) {
    const int lane = threadIdx.x & 31;
    const int wid  = threadIdx.x >> 5;
    const int half = lane >> 4;     // 0: K=k,k+1   1: K=k+2,k+3
    const int l16  = lane & 15;
    const int arow  = blockIdx.x * 128 + wid * 16 + l16;
    const int nbase = blockIdx.y * 64;

    v8f acc[4] = {};
    const float* Arow = A + (size_t)arow * lda;

    for (int k = 0; k < K; k += 8) {
#pragma unroll
        for (int u = 0; u < 2; ++u) {
            const int kk = k + u * 4 + half * 2;
            v2f a = *(const v2f*)(Arow + kk);
#pragma unroll
            for (int j = 0; j < 4; ++j) {
                const int col = nbase + j * 16 + l16;
                v2f b;
                if (TRANSB) {
                    b = *(const v2f*)(Bm + (size_t)col * ldb + kk);
                } else {
                    b.x = Bm[(size_t)kk * ldb + col];
                    b.y = Bm[(size_t)(kk + 1) * ldb + col];
                }
                acc[j] = __builtin_amdgcn_wmma_f32_16x16x4_f32(
                    false, a, false, b, (short)0, acc[j], false, false);
            }
        }
    }

    // C/D layout: VGPR r -> row r (lanes 0-15) / row r+8 (lanes 16-31)
    const int crowbase = blockIdx.x * 128 + wid * 16 + half * 8;
#pragma unroll
    for (int j = 0; j < 4; ++j) {
        const int col = nbase + j * 16 + l16;
        const float bb = bias ? bias[col] : 0.0f;
#pragma unroll
        for (int r = 0; r < 8; ++r) {
            C[(size_t)(crowbase + r) * ldc + col] = acc[j][r] + bb;
        }
    }
}

// ---------------------------------------------------------------------------
// s0[bq,h] = q_h . bk_h  (raw, pre-scale); padded heads 8..15 get 0
// ---------------------------------------------------------------------------
__global__ void s0_kernel(const float* __restrict__ qp,
                          const float* __restrict__ bk,
                          float* __restrict__ s0) {
    int t = blockIdx.x * blockDim.x + threadIdx.x;
    if (t >= BATCH * NQ * 16) return;
    int bq = t >> 4, h = t & 15;
    float acc = 0.0f;
    if (h < NHEADS) {
        const float* qh = qp + (size_t)bq * DIM + h * HD;
        const float* bh = bk + h * HD;
#pragma unroll 8
        for (int i = 0; i < HD; ++i) acc += qh[i] * bh[i];
    }
    s0[t] = acc;
}

// ---------------------------------------------------------------------------
// Window attention core. One workgroup (8 waves) per (b, query).
//   pass 1: S[16h x 512k] = G[16h x 512d] @ Tok^T       (WMMA, mask-zeroed)
//   softmax rows (joint over all 4 levels == reference online merge)
//   pass 2: tagg[16h x 512d] = W[16h x 512k] @ Tok      (WMMA)
// ---------------------------------------------------------------------------
__global__ void attn_kernel(const float* __restrict__ tcenter,
                            const float* __restrict__ fcenter,
                            const float* __restrict__ value,
                            const float* __restrict__ G,
                            const float* __restrict__ s0,
                            float* __restrict__ tagg) {
    __shared__ int   idx_s[NKEYS];
    __shared__ float msk_s[NKEYS];
    __shared__ __align__(16) float S[16 * SSTRIDE];

    const int bq   = blockIdx.x;
    const int b    = bq >> 9;               // / NQ
    const int tid  = threadIdx.x;
    const int lane = tid & 31;
    const int wid  = tid >> 5;
    const int half = lane >> 4;
    const int l16  = lane & 15;

    // ---- window index / OOB mask setup -----------------------------------
    const float tc = tcenter[bq];
    const float fc = fcenter[bq];
    const int Hs[4]  = {64, 32, 16, 8};
    const int Ws[4]  = {1024, 512, 256, 128};
    const int LSI[4] = {0, 65536, 81920, 86016};
    for (int kk = tid; kk < NKEYS; kk += 256) {
        const int lid    = kk >> 7;
        const int within = kk & 127;
        const int t = within & 15;          // 0..15 (wt = 16)
        const int f = within >> 4;          // 0..7  (wf = 8)
        const int W = Ws[lid], H = Hs[lid];
        const int tpx = (int)rintf(tc * (float)W - 0.5f);
        const int fpx = (int)rintf(fc * (float)H - 0.5f);
        const int tt = tpx + t - 8;         // wt/2 = 8
        const int ff = fpx + f - 4;         // wf/2 = 4
        const bool oob = (tt < 0) || (tt >= W) || (ff < 0) || (ff >= H);
        const int ttc = min(max(tt, 0), W - 1);
        const int ffc = min(max(ff, 0), H - 1);
        idx_s[kk] = LSI[lid] + ffc * W + ttc;
        msk_s[kk] = oob ? 0.0f : 1.0f;
    }
    __syncthreads();

    const float* vb   = value + (size_t)b * NTOK * DIM;
    const float* Grow = G + ((size_t)bq * 16 + l16) * DIM;

    // ---- pass 1: scores ---------------------------------------------------
    {
        v8f acc[4] = {};
        const float* tb[4];
        float m[4];
#pragma unroll
        for (int j = 0; j < 4; ++j) {
            const int key = wid * 64 + j * 16 + l16;
            tb[j] = vb + (size_t)idx_s[key] * DIM;
            m[j]  = msk_s[key];
        }
        for (int d = 0; d < DIM; d += 8) {
#pragma unroll
            for (int u = 0; u < 2; ++u) {
                const int kk = d + u * 4 + half * 2;
                v2f a = *(const v2f*)(Grow + kk);
#pragma unroll
                for (int j = 0; j < 4; ++j) {
                    v2f bf = *(const v2f*)(tb[j] + kk);
                    bf.x *= m[j];
                    bf.y *= m[j];
                    acc[j] = __builtin_amdgcn_wmma_f32_16x16x4_f32(
                        false, a, false, bf, (short)0, acc[j], false, false);
                }
            }
        }
        // score = (tok.g + q.bk) / sqrt(hd);  sqrt(64) = 8
        float s0v[8];
#pragma unroll
        for (int r = 0; r < 8; ++r) s0v[r] = s0[bq * 16 + r + half * 8];
#pragma unroll
        for (int j = 0; j < 4; ++j) {
            const int col = wid * 64 + j * 16 + l16;
#pragma unroll
            for (int r = 0; r < 8; ++r) {
                const int rowh = r + half * 8;
                S[rowh * SSTRIDE + col] = (acc[j][r] + s0v[r]) * 0.125f;
            }
        }
    }
    __syncthreads();

    // ---- softmax over all 512 keys (rows wid and wid+8 per wave) ---------
    for (int rr = 0; rr < 2; ++rr) {
        const int row = wid + rr * 8;
        float v[16];
        float mx = -3.4e38f;
#pragma unroll
        for (int t = 0; t < 16; ++t) {
            v[t] = S[row * SSTRIDE + lane + t * 32];
            mx = fmaxf(mx, v[t]);
        }
#pragma unroll
        for (int o = 16; o >= 1; o >>= 1) mx = fmaxf(mx, __shfl_xor(mx, o, 32));
        float sum = 0.0f;
#pragma unroll
        for (int t = 0; t < 16; ++t) { v[t] = expf(v[t] - mx); sum += v[t]; }
#pragma unroll
        for (int o = 16; o >= 1; o >>= 1) sum += __shfl_xor(sum, o, 32);
        const float inv = 1.0f / sum;
#pragma unroll
        for (int t = 0; t < 16; ++t) S[row * SSTRIDE + lane + t * 32] = v[t] * inv;
    }
    __syncthreads();

    // ---- pass 2: weighted token aggregation ------------------------------
    {
        v8f acc[4] = {};
        for (int k0 = 0; k0 < NKEYS; k0 += 8) {
#pragma unroll
            for (int u = 0; u < 2; ++u) {
                const int keyA = k0 + u * 4 + half * 2;
                const int keyB = keyA + 1;
                const int   iA = idx_s[keyA];
                const int   iB = idx_s[keyB];
                const float mA = msk_s[keyA];
                const float mB = msk_s[keyB];
                const float* ta = vb + (size_t)iA * DIM;
                const float* tb2 = vb + (size_t)iB * DIM;
                // A-frag: softmax weights from LDS, row = l16, K = keyA,keyA+1
                v2f a = *(const v2f*)(&S[l16 * SSTRIDE + keyA]);
#pragma unroll
                for (int j = 0; j < 4; ++j) {
                    const int dim = wid * 64 + j * 16 + l16;
                    v2f bf;
                    bf.x = ta[dim]  * mA;
                    bf.y = tb2[dim] * mB;
                    acc[j] = __builtin_amdgcn_wmma_f32_16x16x4_f32(
                        false, a, false, bf, (short)0, acc[j], false, false);
                }
            }
        }
        // rows 0..7 (real heads) live in lanes 0-15 components r=0..7
        if (half == 0) {
#pragma unroll
            for (int j = 0; j < 4; ++j) {
                const int dim = wid * 64 + j * 16 + l16;
#pragma unroll
                for (int r = 0; r < 8; ++r) {
                    tagg[((size_t)bq * NHEADS + r) * DIM + dim] = acc[j][r];
                }
            }
        }
    }
}

// ---------------------------------------------------------------------------
// Host-side orchestration
// ---------------------------------------------------------------------------
extern "C" void kernel_launch(void* const* d_in, const int* in_sizes, int n_in,
                              void* d_out, int out_size, void* d_ws, size_t ws_size,
                              hipStream_t stream) {
    const float* query  = (const float*)d_in[0];
    const float* tcent  = (const float*)d_in[1];
    const float* fcent  = (const float*)d_in[2];
    const float* value  = (const float*)d_in[3];
    const float* Wq     = (const float*)d_in[4];
    const float* bq     = (const float*)d_in[5];
    const float* Wk     = (const float*)d_in[6];
    const float* bk     = (const float*)d_in[7];
    const float* Wv     = (const float*)d_in[8];
    const float* bv     = (const float*)d_in[9];
    const float* Wo     = (const float*)d_in[10];
    const float* bo     = (const float*)d_in[11];
    float* out = (float*)d_out;

    const int BQ = BATCH * NQ;                 // 1024
    float* ws   = (float*)d_ws;
    float* qp   = ws;                          // [BQ, 512]
    float* Gbuf = qp + (size_t)BQ * DIM;       // [BQ, 16, 512]  (padded heads)
    float* s0   = Gbuf + (size_t)BQ * 16 * DIM;// [BQ, 16]
    float* tagg = s0 + (size_t)BQ * 16;        // [BQ, 8, 512]
    float* Obuf = tagg + (size_t)BQ * NHEADS * DIM; // [BQ, 512]

    const dim3 blk(256);

    // 0) zero padded-G rows (8..15 stay zero -> benign padded WMMA rows)
    {
        const int n = BQ * 16 * DIM;
        fill_zero_kernel<<<(n + 255) / 256, blk, 0, stream>>>(Gbuf, n);
    }

    // 1) qp = query @ Wq + bq            [1024,512] x [512,512]
    gemm_wmma_f32<0><<<dim3(BQ / 128, DIM / 64), blk, 0, stream>>>(
        query, DIM, Wq, DIM, bq, qp, DIM, DIM);

    // 2) s0[bq,h] = q_h . bk_h
    s0_kernel<<<(BQ * 16 + 255) / 256, blk, 0, stream>>>(qp, bk, s0);

    // 3) G[bq,h,d] = sum_c qp[bq,h*64+c] * Wk[d, h*64+c]   (B transposed)
    for (int h = 0; h < NHEADS; ++h) {
        gemm_wmma_f32<1><<<dim3(BQ / 128, DIM / 64), blk, 0, stream>>>(
            qp + h * HD, DIM, Wk + h * HD, DIM, nullptr,
            Gbuf + h * DIM, 16 * DIM, HD);
    }

    // 4) attention: scores -> softmax -> weighted raw-token aggregation
    attn_kernel<<<dim3(BQ), blk, 0, stream>>>(tcent, fcent, value, Gbuf, s0, tagg);

    // 5) O[bq, h*64+n] = tagg[bq,h,:] @ Wv[:, h*64+n] + bv[h*64+n]
    for (int h = 0; h < NHEADS; ++h) {
        gemm_wmma_f32<0><<<dim3(BQ / 128, 1), blk, 0, stream>>>(
            tagg + h * DIM, NHEADS * DIM, Wv + h * HD, DIM, bv + h * HD,
            Obuf + h * HD, DIM, DIM);
    }

    // 6) out = O @ Wo + bo
    gemm_wmma_f32<0><<<dim3(BQ / 128, DIM / 64), blk, 0, stream>>>(
        Obuf, DIM, Wo, DIM, bo, out, DIM, DIM);
}